// DeepAR_31061203485038
// MI455X (gfx1250) — compile-verified
//
#include <hip/hip_runtime.h>
#include <hip/hip_bf16.h>
#include <math.h>

typedef __attribute__((ext_vector_type(2))) float v2f;
typedef __attribute__((ext_vector_type(8))) float v8f;

#define B_TOT   4096
#define S_LEN   512
#define HIDN    64
#define G4      256     // 4*HID
#define NCOV    10
#define EMB     16
#define TB      16      // batch rows per workgroup
#define HPITCH  68      // LDS pitch for h (bank-conflict-free A-tile loads)
#define SGPITCH 260     // LDS pitch for static-gate buffer
#define XPITCH  12      // dyn input padded to K=12 (3 K-tiles of 4)
#define CHUNK   16      // timesteps staged per LDS refill
#define CPITCH  (CHUNK*NCOV)

struct __align__(16) SMem {
  float covS[TB * CPITCH];  // staged covariates (float4 loads)
  float sg[TB * SGPITCH];   // static gate contribution (+bias), read-only after init
  float h[TB * HPITCH];     // hidden state (A operand of recurrent WMMA)
  float xdyn[TB * XPITCH];  // [prev_scaled, cov(10), 0]
  float prevS[TB * CHUNK];  // target[t-1] staged (0 at t==0)
  float sx[TB * 65];        // static_x
  float part[4 * 32];       // output-head partial sums
  float wo[HIDN * 2];
  float bo[2];
  float scl[TB];
  float iscl[TB];
};

__device__ __forceinline__ float fast_tanh(float x) {
#if __has_builtin(__builtin_amdgcn_tanhf)
  return __builtin_amdgcn_tanhf(x);           // v_tanh_f32 (CDNA5 TRANS op)
#else
  float y;
  asm("v_tanh_f32 %0, %1" : "=v"(y) : "v"(x));
  return y;
#endif
}

__device__ __forceinline__ float fast_sigmoid(float x) {
  return __builtin_amdgcn_rcpf(1.0f + __expf(-x));  // v_exp + v_rcp
}

__global__ __launch_bounds__(128, 1)
void deepar_lstm_kernel(const float* __restrict__ target,
                        const float* __restrict__ covs,
                        const int*   __restrict__ cats,
                        const float* __restrict__ scale,
                        const float* __restrict__ emb0,
                        const float* __restrict__ emb1,
                        const float* __restrict__ emb2,
                        const float* __restrict__ emb3,
                        const float* __restrict__ w_ih,
                        const float* __restrict__ w_hh,
                        const float* __restrict__ bias,
                        const float* __restrict__ w_out,
                        const float* __restrict__ b_out,
                        float* __restrict__ out)
{
  __shared__ SMem sm;
  const int tid  = threadIdx.x;
  const int wave = tid >> 5;
  const int lane = tid & 31;
  const int b0   = blockIdx.x * TB;

  // ---------------- one-time setup ----------------
  if (tid < TB) {
    float s = scale[b0 + tid];
    sm.scl[tid]  = s;
    sm.iscl[tid] = 1.0f / fmaxf(s, 1e-4f);
  }
  if (tid < HIDN * 2) sm.wo[tid] = w_out[tid];
  if (tid < 2)        sm.bo[tid] = b_out[tid];

  for (int e = tid; e < TB * 65; e += 128) {
    int m = e / 65, k = e % 65;
    float v;
    if      (k < 16) v = emb0[cats[(b0 + m) * 4 + 0] * EMB + k];
    else if (k < 32) v = emb1[cats[(b0 + m) * 4 + 1] * EMB + (k - 16)];
    else if (k < 48) v = emb2[cats[(b0 + m) * 4 + 2] * EMB + (k - 32)];
    else if (k < 64) v = emb3[cats[(b0 + m) * 4 + 3] * EMB + (k - 48)];
    else             v = log1pf(scale[b0 + m]);
    sm.sx[e] = v;
  }
  __syncthreads();

  // static gate precompute: sg[m][j] = bias[j] + static_x[m] . w_ih[11:76, j]
  for (int e = tid; e < TB * G4; e += 128) {
    int m = e >> 8, j = e & 255;
    float acc = bias[j];
    #pragma unroll 5
    for (int k = 0; k < 65; ++k)
      acc += sm.sx[m * 65 + k] * w_ih[(11 + k) * G4 + j];
    sm.sg[m * SGPITCH + j] = acc;
  }

  // ---------------- per-wave register tiles ----------------
  // wave w owns columns {nt*64 + w*16 .. +15} of the gate matrix, nt = i,f,g,o
  const int khalf = lane >> 4;       // f32 WMMA A/B layout: 0 -> K=0,1  1 -> K=2,3
  const int ml    = lane & 15;
  const int ncol  = wave * 16 + ml;

  v2f whhB[16][4];                   // B operands of h @ w_hh (K=64 -> 16 K-tiles)
  #pragma unroll
  for (int kt = 0; kt < 16; ++kt)
    #pragma unroll
    for (int nt = 0; nt < 4; ++nt) {
      int k = kt * 4 + khalf * 2;
      int n = nt * 64 + ncol;
      v2f b = { w_hh[k * G4 + n], w_hh[(k + 1) * G4 + n] };
      whhB[kt][nt] = b;
    }
  v2f wihB[3][4];                    // B operands of xdyn @ w_ih[0:12] (col 11 hits zero pad)
  #pragma unroll
  for (int kt = 0; kt < 3; ++kt)
    #pragma unroll
    for (int nt = 0; nt < 4; ++nt) {
      int k = kt * 4 + khalf * 2;
      int n = nt * 64 + ncol;
      v2f b = { w_ih[k * G4 + n], w_ih[(k + 1) * G4 + n] };
      wihB[kt][nt] = b;
    }
  __syncthreads();                   // sg complete
  v8f statC[4];                      // static gate contribution in C layout
  #pragma unroll
  for (int nt = 0; nt < 4; ++nt)
    #pragma unroll
    for (int r = 0; r < 8; ++r) {
      int m = r + khalf * 8;
      statC[nt][r] = sm.sg[m * SGPITCH + nt * 64 + ncol];
    }

  // zero h, stage chunk 0, build xdyn(0)
  for (int e = tid; e < TB * HPITCH; e += 128) sm.h[e] = 0.0f;
  for (int e4 = tid; e4 < TB * (CPITCH / 4); e4 += 128) {
    int m = e4 / (CPITCH / 4), off4 = e4 % (CPITCH / 4);
    *(float4*)&sm.covS[m * CPITCH + off4 * 4] =
        *(const float4*)&covs[(size_t)(b0 + m) * S_LEN * NCOV + off4 * 4];
    if ((off4 & 7) == 0)
      __builtin_prefetch(&covs[((size_t)(b0 + m) * S_LEN + CHUNK) * NCOV + off4 * 4], 0, 1);
  }
  for (int e = tid; e < TB * CHUNK; e += 128) {
    int m = e >> 4, j = e & 15;
    sm.prevS[e] = (j == 0) ? 0.0f : target[(size_t)(b0 + m) * S_LEN + j - 1];
  }
  __syncthreads();
  for (int e = tid; e < TB * XPITCH; e += 128) {
    int m = e / XPITCH, k = e % XPITCH;
    float v;
    if      (k == 0)    v = sm.prevS[m * CHUNK] * sm.iscl[m];
    else if (k <= NCOV) v = sm.covS[m * CPITCH + (k - 1)];
    else                v = 0.0f;
    sm.xdyn[e] = v;
  }

  v8f creg = {0.f, 0.f, 0.f, 0.f, 0.f, 0.f, 0.f, 0.f};
  v8f hreg;

  // ---------------- sequential scan ----------------
  for (int t = 0; t < S_LEN; ++t) {
    __syncthreads();   // barrier A: xdyn(t), h(t-1), staged chunk visible

    // ---- read region: WMMA + in-register LSTM cell ----
    v2f xA[3], hA[16];
    #pragma unroll
    for (int kt = 0; kt < 3; ++kt)
      xA[kt] = *(const v2f*)&sm.xdyn[ml * XPITCH + kt * 4 + khalf * 2];
    #pragma unroll
    for (int kt = 0; kt < 16; ++kt)
      hA[kt] = *(const v2f*)&sm.h[ml * HPITCH + kt * 4 + khalf * 2];

    v8f acc[4];
    #pragma unroll
    for (int nt = 0; nt < 4; ++nt) {
      v8f a = statC[nt];
      #pragma unroll
      for (int kt = 0; kt < 3; ++kt)
        a = __builtin_amdgcn_wmma_f32_16x16x4_f32(false, xA[kt], false, wihB[kt][nt],
                                                  (short)0, a, false, false);
      #pragma unroll
      for (int kt = 0; kt < 16; ++kt)
        a = __builtin_amdgcn_wmma_f32_16x16x4_f32(false, hA[kt], false, whhB[kt][nt],
                                                  (short)0, a, false, false);
      acc[nt] = a;
    }
    // i,f,o -> sigmoid ; g -> tanh ; cell update entirely in registers (C layout)
    #pragma unroll
    for (int r = 0; r < 8; ++r) {
      float gi = fast_sigmoid(acc[0][r]);
      float gf = fast_sigmoid(acc[1][r]);
      float gg = fast_tanh(acc[2][r]);
      float go = fast_sigmoid(acc[3][r]);
      float cc = gf * creg[r] + gi * gg;
      creg[r] = cc;
      hreg[r] = go * fast_tanh(cc);
    }
    // output-head partials for h(t-1) (all 128 threads; combined next region)
    {
      int p = tid & 31, q = tid >> 5;
      int m = p >> 1, j = p & 1;
      float s = 0.0f;
      #pragma unroll
      for (int kk = 0; kk < 16; ++kk) {
        int k = q * 16 + kk;
        s += sm.h[m * HPITCH + k] * sm.wo[k * 2 + j];
      }
      sm.part[q * 32 + p] = s;
    }

    __syncthreads();   // barrier B: all reads of h(t-1)/xdyn(t) done

    // ---- write region: commit h(t), emit output(t-1), stage next step ----
    #pragma unroll
    for (int r = 0; r < 8; ++r) {
      int m = r + khalf * 8;
      sm.h[m * HPITCH + ncol] = hreg[r];
    }
    if (t > 0 && tid < 32) {
      int m = tid >> 1, j = tid & 1;
      float r = sm.bo[j] + sm.part[tid] + sm.part[32 + tid] +
                sm.part[64 + tid] + sm.part[96 + tid];
      float sp = (r > 20.0f) ? r : log1pf(__expf(r));
      sp += 1e-4f;
      size_t idx = (size_t)(b0 + m) * S_LEN + (t - 1);
      if (j == 0) out[idx] = sp * sm.scl[m];
      else        out[(size_t)B_TOT * S_LEN + idx] = sp;
    }
    const int tn = t + 1;
    if (tn < S_LEN) {
      if ((tn & (CHUNK - 1)) == 0) {
        // stage next 16-step chunk (b128 loads) + prefetch the one after
        for (int e4 = tid; e4 < TB * (CPITCH / 4); e4 += 128) {
          int m = e4 / (CPITCH / 4), off4 = e4 % (CPITCH / 4);
          *(float4*)&sm.covS[m * CPITCH + off4 * 4] =
              *(const float4*)&covs[((size_t)(b0 + m) * S_LEN + tn) * NCOV + off4 * 4];
          if ((off4 & 7) == 0 && tn + CHUNK < S_LEN)
            __builtin_prefetch(&covs[((size_t)(b0 + m) * S_LEN + tn + CHUNK) * NCOV + off4 * 4], 0, 1);
        }
        for (int e = tid; e < TB * CHUNK; e += 128) {
          int m = e >> 4, j = e & 15;
          sm.prevS[e] = target[(size_t)(b0 + m) * S_LEN + tn + j - 1];
        }
        __syncthreads();
      }
      const int tc = tn & (CHUNK - 1);
      for (int e = tid; e < TB * XPITCH; e += 128) {
        int m = e / XPITCH, k = e % XPITCH;
        float v;
        if      (k == 0)    v = sm.prevS[m * CHUNK + tc] * sm.iscl[m];
        else if (k <= NCOV) v = sm.covS[m * CPITCH + tc * NCOV + (k - 1)];
        else                v = 0.0f;
        sm.xdyn[e] = v;
      }
    }
  }

  // ---------------- epilogue: output for t = S_LEN-1 ----------------
  __syncthreads();
  {
    int p = tid & 31, q = tid >> 5;
    int m = p >> 1, j = p & 1;
    float s = 0.0f;
    #pragma unroll
    for (int kk = 0; kk < 16; ++kk) {
      int k = q * 16 + kk;
      s += sm.h[m * HPITCH + k] * sm.wo[k * 2 + j];
    }
    sm.part[q * 32 + p] = s;
  }
  __syncthreads();
  if (tid < 32) {
    int m = tid >> 1, j = tid & 1;
    float r = sm.bo[j] + sm.part[tid] + sm.part[32 + tid] +
              sm.part[64 + tid] + sm.part[96 + tid];
    float sp = (r > 20.0f) ? r : log1pf(__expf(r));
    sp += 1e-4f;
    size_t idx = (size_t)(b0 + m) * S_LEN + (S_LEN - 1);
    if (j == 0) out[idx] = sp * sm.scl[m];
    else        out[(size_t)B_TOT * S_LEN + idx] = sp;
  }
}

extern "C" void kernel_launch(void* const* d_in, const int* in_sizes, int n_in,
                              void* d_out, int out_size, void* d_ws, size_t ws_size,
                              hipStream_t stream) {
  const float* target = (const float*)d_in[0];
  const float* covs   = (const float*)d_in[1];
  const int*   cats   = (const int*)  d_in[2];
  const float* scale  = (const float*)d_in[3];
  const float* emb0   = (const float*)d_in[4];
  const float* emb1   = (const float*)d_in[5];
  const float* emb2   = (const float*)d_in[6];
  const float* emb3   = (const float*)d_in[7];
  const float* w_ih   = (const float*)d_in[8];
  const float* w_hh   = (const float*)d_in[9];
  const float* bias   = (const float*)d_in[10];
  const float* w_out  = (const float*)d_in[11];
  const float* b_out  = (const float*)d_in[12];
  float* out = (float*)d_out;

  dim3 grid(B_TOT / TB);   // 256 workgroups, one 16-row batch tile each
  dim3 block(128);         // 4 wave32s; each owns a 16-col slice of all 4 gates
  deepar_lstm_kernel<<<grid, block, 0, stream>>>(
      target, covs, cats, scale, emb0, emb1, emb2, emb3,
      w_ih, w_hh, bias, w_out, b_out, out);
}